// AdaptiveSpectralFeatureRefinementCosine_72739566125923
// MI455X (gfx1250) — compile-verified
//
#include <hip/hip_runtime.h>
#include <stdint.h>

// Problem constants (B,C,H,W) = (4,128,128,128), p=3, K=9
#define Bdim 4
#define Cdim 128
#define Hdim 128
#define Wdim 128
#define LDSW (Wdim + 8)   // x stored at [x+4]; zero pads give free boundary handling

typedef __attribute__((ext_vector_type(2))) float v2f;
typedef __attribute__((ext_vector_type(8))) float v8f;

#if __has_builtin(__builtin_amdgcn_wmma_f32_16x16x4_f32)
#define HAVE_WMMA_F32_4 1
#else
#define HAVE_WMMA_F32_4 0
#warning "CDNA5: wmma_f32_16x16x4_f32 builtin NOT available - VALU fallback"
#endif

// ---- CDNA5 async global->LDS (per-lane, ASYNCcnt) ----
#if defined(__gfx1250__) && __has_builtin(__builtin_amdgcn_global_load_async_to_lds_b128)
#define HAVE_ASYNC_B128 1
typedef int v4i_vs __attribute__((vector_size(16)));
typedef __attribute__((address_space(1))) v4i_vs* gptr_v4i;
typedef __attribute__((address_space(3))) v4i_vs* lptr_v4i;
__device__ __forceinline__ void async_copy_b128(const float* g, float* l) {
  __builtin_amdgcn_global_load_async_to_lds_b128((gptr_v4i)(void*)g,
                                                 (lptr_v4i)(void*)l, 0, 0);
}
__device__ __forceinline__ void wait_asynccnt0() {
#if __has_builtin(__builtin_amdgcn_s_wait_asynccnt)
  __builtin_amdgcn_s_wait_asynccnt(0);
#else
  asm volatile("s_wait_asynccnt 0" ::: "memory");
#endif
}
#else
#define HAVE_ASYNC_B128 0
#warning "CDNA5: global_load_async_to_lds_b128 builtin NOT available"
#endif

// ---- CDNA5 Tensor Data Mover (TDM, TENSORcnt) ----
#if defined(__gfx1250__) && __has_builtin(__builtin_amdgcn_tensor_load_to_lds)
#define HAVE_TDM 1
typedef __attribute__((ext_vector_type(4))) unsigned tdm_u32x4;
typedef __attribute__((ext_vector_type(8))) int tdm_i32x8;
typedef __attribute__((ext_vector_type(4))) int tdm_i32x4;

__device__ __forceinline__ unsigned lds_offset_of(const void* p) {
  return (unsigned)(size_t)(__attribute__((address_space(3))) const void*)p;
}

// Load a 2D tile: 128 contiguous f32 per line, 128 lines, line stride H*W f32.
// TDM pad feature inserts 8 dwords after every 128 dwords -> LDS row stride
// 136 f32, matching the bank-conflict-free layout used by the WMMA fragments.
__device__ __forceinline__ void tdm_load_rows_128x128(unsigned lds_byte,
                                                      const float* gsrc) {
  const uint64_t ga = (uint64_t)(uintptr_t)gsrc;
  tdm_u32x4 g0;
  g0.x = 1u;                                                  // count=1, user D#
  g0.y = lds_byte;                                            // lds_addr (bytes)
  g0.z = (unsigned)ga;                                        // global_addr[31:0]
  g0.w = (unsigned)((ga >> 32) & 0x01FFFFFFu) | (2u << 30);   // addr[56:32]|type=2
  tdm_i32x8 g1;
  g1[0] = (int)((2u << 16)      // data_size = 4B
              | (1u << 20)      // pad_enable
              | (6u << 22)      // pad_interval: 128 dwords
              | (7u << 25));    // pad_amount: 8 dwords
  g1[1] = (int)(128u << 16);    // tensor_dim0 = 128 (low 16 of bits 79:48)
  g1[2] = (int)(128u << 16);    // tensor_dim0 hi=0 | tensor_dim1 = 128 (low 16)
  g1[3] = (int)(128u << 16);    // tensor_dim1 hi=0 | tile_dim0 = 128
  g1[4] = (int)(128u);          // tile_dim1 = 128 | tile_dim2 = 0
  g1[5] = (int)(Hdim * Wdim);   // tensor_dim0_stride = 16384 (low 32)
  g1[6] = 0;                    // stride hi | dim1_stride (unused, 2D)
  g1[7] = 0;
  tdm_i32x4 g2 = {0, 0, 0, 0};
  tdm_i32x4 g3 = {0, 0, 0, 0};
#if __has_include(<hip/amd_detail/amd_gfx1250_TDM.h>)
  tdm_i32x8 g4 = {};
  __builtin_amdgcn_tensor_load_to_lds(g0, g1, g2, g3, g4, 0);  // clang-23 arity
#else
  __builtin_amdgcn_tensor_load_to_lds(g0, g1, g2, g3, 0);      // ROCm 7.2 arity
#endif
}
__device__ __forceinline__ void wait_tensorcnt0() {
#if __has_builtin(__builtin_amdgcn_s_wait_tensorcnt)
  __builtin_amdgcn_s_wait_tensorcnt(0);
#else
  asm volatile("s_wait_tensorcnt 0" ::: "memory");
#endif
}
#else
#define HAVE_TDM 0
#warning "CDNA5: tensor_load_to_lds builtin NOT available"
#endif

__global__ __launch_bounds__(256) void asfr_cosine_refine_kernel(
    const float* __restrict__ fe, const float* __restrict__ fu,
    float* __restrict__ out) {
  const int t = threadIdx.x;
  const int b = blockIdx.x >> 7;   // B*H = 512 blocks
  const int y = blockIdx.x & 127;

  // ~288 KB static LDS — CDNA5 WGP has 320 KB
  __shared__ __align__(16) float s_fu[3][Cdim][LDSW];  // raw fused rows y-1..y+1
  __shared__ __align__(16) float s_fe[Cdim][LDSW];     // raw fe_lv row y (x at [x])
  __shared__ float s_D[8][16][16];                     // per-wave WMMA D staging
  __shared__ float s_cos[Wdim][12];                    // cos then softmax weights
  __shared__ float s_invP[3][LDSW];
  __shared__ float s_invF[Wdim];

  // ---- zero fill padded fused buffer (x and y boundary padding) ----
  {
    float4 z4 = make_float4(0.f, 0.f, 0.f, 0.f);
    float4* z = (float4*)&s_fu[0][0][0];
    const int n4 = 3 * Cdim * LDSW / 4;
    for (int i = t; i < n4; i += 256) z[i] = z4;
  }
  __syncthreads();

  // ---- stream global -> LDS on the CDNA5 DMA paths ----
#if HAVE_TDM
  if (t == 0) {  // TDM ignores EXEC; issue one descriptor per valid fused row
    for (int dy = 0; dy < 3; ++dy) {
      const int yy = y + dy - 1;
      if (yy < 0 || yy >= Hdim) continue;
      tdm_load_rows_128x128(lds_offset_of(&s_fu[dy][0][4]),
                            fu + ((size_t)b * Cdim * Hdim + yy) * (size_t)Wdim);
    }
  }
#else
  for (int dy = 0; dy < 3; ++dy) {
    const int yy = y + dy - 1;
    if (yy < 0 || yy >= Hdim) continue;
    for (int idx = t; idx < Cdim * (Wdim / 4); idx += 256) {
      const int c = idx >> 5, q = idx & 31;
      const float4* src = (const float4*)(fu + (((size_t)b * Cdim + c) * Hdim + yy) * Wdim);
      *(float4*)&s_fu[dy][c][4 * q + 4] = src[q];
    }
  }
#endif

#if HAVE_ASYNC_B128
  for (int idx = t; idx < Cdim * (Wdim / 4); idx += 256) {
    const int c = idx >> 5, q = idx & 31;
    async_copy_b128(fe + (((size_t)b * Cdim + c) * Hdim + y) * Wdim + 4 * q,
                    &s_fe[c][4 * q]);
  }
  wait_asynccnt0();
#else
  for (int idx = t; idx < Cdim * (Wdim / 4); idx += 256) {
    const int c = idx >> 5, q = idx & 31;
    const float4* src = (const float4*)(fe + (((size_t)b * Cdim + c) * Hdim + y) * Wdim);
    *(float4*)&s_fe[c][4 * q] = src[q];
  }
#endif
#if HAVE_TDM
  wait_tensorcnt0();
#endif
  __syncthreads();

  // ---- per-column inverse L2 norms (factored out of the dots) ----
  for (int idx = t; idx < 3 * LDSW; idx += 256) {
    const int dy = idx / LDSW, xp = idx % LDSW;
    float ss = 0.f;
    for (int c = 0; c < Cdim; ++c) { float v = s_fu[dy][c][xp]; ss += v * v; }
    s_invP[dy][xp] = 1.f / fmaxf(sqrtf(ss), 1e-12f);
  }
  if (t < Wdim) {
    float ss = 0.f;
    for (int c = 0; c < Cdim; ++c) { float v = s_fe[c][t]; ss += v * v; }
    s_invF[t] = 1.f / fmaxf(sqrtf(ss), 1e-12f);
  }
  __syncthreads();

  // ---- banded GEMM via V_WMMA_F32_16X16X4_F32: each wave owns 16 pixels ----
  const int wv = t >> 5;
  const int lane = t & 31;
  const int half = lane >> 4;   // A/B lane layout: lanes 16-31 hold K=2,3
  const int lm = lane & 15;
  const int x0 = wv * 16;

#if HAVE_WMMA_F32_4
  // Preload A fragments once (reused across the 3 dy GEMMs). 64 VGPRs.
  float a0[32], a1[32];
#pragma unroll
  for (int s = 0; s < 32; ++s) {
    const int c = 4 * s + 2 * half;
    a0[s] = s_fe[c][x0 + lm];
    a1[s] = s_fe[c + 1][x0 + lm];
  }
#endif

  for (int dy = 0; dy < 3; ++dy) {
#if HAVE_WMMA_F32_4
    v8f d = {0.f, 0.f, 0.f, 0.f, 0.f, 0.f, 0.f, 0.f};
#pragma unroll
    for (int s = 0; s < 32; ++s) {
      const int c = 4 * s + 2 * half;
      v2f A; A.x = a0[s]; A.y = a1[s];
      // B col n -> global x = x0 + n - 1 -> padded idx x0 + n + 3
      v2f Bv; Bv.x = s_fu[dy][c][x0 + lm + 3]; Bv.y = s_fu[dy][c + 1][x0 + lm + 3];
      d = __builtin_amdgcn_wmma_f32_16x16x4_f32(false, A, false, Bv, (short)0, d,
                                                false, false);
    }
#pragma unroll
    for (int r = 0; r < 8; ++r) s_D[wv][r + 8 * half][lm] = d[r];
#endif
    __syncthreads();
    if (lane < 16) {
      const int x = x0 + lm;
      const float invf = s_invF[x];
#pragma unroll
      for (int dx = -1; dx <= 1; ++dx) {
        const int j = lm + dx + 1;   // D[i][j] pairs pixel i with x+dx
        float dot;
#if HAVE_WMMA_F32_4
        if (j < 16) {
          dot = s_D[wv][lm][j];
        } else
#endif
        {
          dot = 0.f;  // tile-edge entries (3 per dy) done directly
          for (int c = 0; c < Cdim; ++c) dot += s_fe[c][x] * s_fu[dy][c][x + dx + 4];
        }
        s_cos[x][3 * dy + dx + 1] = dot * invf * s_invP[dy][x + dx + 4];
      }
    }
    __syncthreads();
  }

  // ---- softmax over the 9 neighbors, per pixel ----
  if (t < Wdim) {
    float c9[9], m = -3.0e38f;
#pragma unroll
    for (int k = 0; k < 9; ++k) { c9[k] = s_cos[t][k]; m = fmaxf(m, c9[k]); }
    float sum = 0.f;
#pragma unroll
    for (int k = 0; k < 9; ++k) { c9[k] = __expf(c9[k] - m); sum += c9[k]; }
    const float inv = 1.f / sum;
#pragma unroll
    for (int k = 0; k < 9; ++k) s_cos[t][k] = c9[k] * inv;
  }
  __syncthreads();

  // ---- weighted aggregation of raw patches + residual, coalesced store ----
  {
    const int x = t & 127;
    const int c0 = t >> 7;
    float w[9];
#pragma unroll
    for (int k = 0; k < 9; ++k) w[k] = s_cos[x][k];
    for (int c = c0; c < Cdim; c += 2) {
      float acc = s_fe[c][x];  // + fe_lv residual
#pragma unroll
      for (int k = 0; k < 9; ++k)
        acc += w[k] * s_fu[k / 3][c][x + (k % 3) + 3];
      out[(((size_t)b * Cdim + c) * Hdim + y) * Wdim + x] = acc;
    }
  }
}

// ===================== harness entry =====================
extern "C" void kernel_launch(void* const* d_in, const int* in_sizes, int n_in,
                              void* d_out, int out_size, void* d_ws, size_t ws_size,
                              hipStream_t stream) {
  (void)in_sizes; (void)n_in; (void)d_ws; (void)ws_size; (void)out_size;
  const float* fe = (const float*)d_in[0];   // fe_lv
  const float* fu = (const float*)d_in[1];   // fused_features
  float* out = (float*)d_out;
  dim3 grid(Bdim * Hdim);   // one workgroup per (batch, row)
  asfr_cosine_refine_kernel<<<grid, 256, 0, stream>>>(fe, fu, out);
}